// Patchifier_48180943127022
// MI455X (gfx1250) — compile-verified
//
#include <hip/hip_runtime.h>

// Problem geometry (fixed by the reference):
//   score_maps [4, 32, 480, 640] f32, GRID=10 -> cells of 48x64, 100 cells/map.
#define MAP_H 480
#define MAP_W 640
#define GRIDN 10
#define CH 48
#define CW 64
#define CELL_ELEMS (CH * CW)        // 3072 floats = 12 KB
#define CELLS_PER_MAP (GRIDN * GRIDN)
#define WAVES_PER_BLOCK 4

// ---------------------------------------------------------------------------
// Kernel 1: one wave32 per 48x64 cell.
//   - 24x global_load_async_to_lds_b128 stream the cell into LDS (row-major,
//     LDS offset == cell flat index * 4), tracked on ASYNCcnt.
//   - split s_wait_asynccnt lets the wave reduce the first half of the cell
//     while the second half is still in flight.
//   - wave32 shfl_xor reduction keeps (max score, min flat index) to match
//     jnp.argmax first-occurrence semantics.
//   - clamped coordinate is always inside the same cell, so the score
//     re-read at the clamped location comes from LDS.
// ---------------------------------------------------------------------------
__global__ void __launch_bounds__(32 * WAVES_PER_BLOCK)
patchifier_cell_argmax(const float* __restrict__ maps,
                       float* __restrict__ candS,
                       int* __restrict__ candR,
                       int* __restrict__ candC,
                       int nCells)
{
    __shared__ __align__(16) float sm[WAVES_PER_BLOCK * CELL_ELEMS]; // 48 KB

    const int lane = threadIdx.x & 31;
    const int wave = threadIdx.x >> 5;
    const int cell = blockIdx.x * WAVES_PER_BLOCK + wave;
    if (cell >= nCells) return;                 // wave-uniform exit

    const int map = cell / CELLS_PER_MAP;
    const int ci  = cell % CELLS_PER_MAP;
    const int gy  = ci / GRIDN;
    const int gx  = ci % GRIDN;

    const float* cellBase =
        maps + (size_t)map * (MAP_H * MAP_W) + (size_t)(gy * CH) * MAP_W + gx * CW;
    float* mySm = &sm[wave * CELL_ELEMS];
    const unsigned ldsBase = (unsigned)(size_t)mySm;  // low 32 bits = LDS offset

    // Issue 24 async b128 loads. Instruction i, lane L covers cell flat floats
    // [i*128 + 4L, i*128 + 4L + 4)  ->  rows 2i + (L>>4), cols (L&15)*4 .. +3.
    // LDS dest offset = flat_index*4, so LDS is an exact row-major cell copy.
#pragma unroll
    for (int i = 0; i < 24; ++i) {
        const int e   = i * 128 + lane * 4;          // flat float index in cell
        const int row = e >> 6;
        const int col = e & 63;
        const unsigned long long ga =
            (unsigned long long)(cellBase + (size_t)row * MAP_W + col);
        const unsigned la = ldsBase + (unsigned)e * 4u;
        asm volatile("global_load_async_to_lds_b128 %0, %1, off"
                     :: "v"(la), "v"(ga) : "memory");
    }

    float best = -__builtin_inff();
    int   bidx = 0;

    // Async loads complete in order: wait until <=12 outstanding -> first 12
    // chunks (k = 0..11) are resident; overlap their reduction with the rest.
    asm volatile("s_wait_asynccnt 0xc" ::: "memory");
#pragma unroll 4
    for (int k = 0; k < 12; ++k) {
        const int e = (k * 32 + lane) * 4;
        const float4 v = *(const float4*)(mySm + e);
        if (v.x > best) { best = v.x; bidx = e; }
        if (v.y > best) { best = v.y; bidx = e + 1; }
        if (v.z > best) { best = v.z; bidx = e + 2; }
        if (v.w > best) { best = v.w; bidx = e + 3; }
    }
    asm volatile("s_wait_asynccnt 0x0" ::: "memory");
#pragma unroll 4
    for (int k = 12; k < 24; ++k) {
        const int e = (k * 32 + lane) * 4;
        const float4 v = *(const float4*)(mySm + e);
        if (v.x > best) { best = v.x; bidx = e; }
        if (v.y > best) { best = v.y; bidx = e + 1; }
        if (v.z > best) { best = v.z; bidx = e + 2; }
        if (v.w > best) { best = v.w; bidx = e + 3; }
    }

    // wave32 reduction: max score, ties -> smaller flat index (first occurrence)
#pragma unroll
    for (int m = 16; m >= 1; m >>= 1) {
        const float os = __shfl_xor(best, m, 32);
        const int   oi = __shfl_xor(bidx, m, 32);
        if (os > best || (os == best && oi < bidx)) { best = os; bidx = oi; }
    }

    if (lane == 0) {
        const int r = bidx >> 6;
        const int c = bidx & 63;
        int gr = gy * CH + r;
        int gc = gx * CW + c;
        gr = (gr < 1) ? 1 : (gr > MAP_H - 2 ? MAP_H - 2 : gr);
        gc = (gc < 1) ? 1 : (gc > MAP_W - 2 ? MAP_W - 2 : gc);
        // Clamped coordinate is always within this cell -> re-read from LDS.
        const float sc = mySm[(gr - gy * CH) * CW + (gc - gx * CW)];
        candS[cell] = sc;
        candR[cell] = gr;
        candC[cell] = gc;
    }
}

// ---------------------------------------------------------------------------
// Kernel 2: one block per map. Stable-descending rank sort of 100 candidates;
// ranks < top_n scatter into d_out = [rows | cols | scores], each [nMaps, top_n].
// Matches jax.lax.top_k ordering (descending, ties -> lower candidate index).
// ---------------------------------------------------------------------------
__global__ void __launch_bounds__(128)
patchifier_topk(const float* __restrict__ candS,
                const int* __restrict__ candR,
                const int* __restrict__ candC,
                const int* __restrict__ topn_ptr,
                float* __restrict__ out,
                int nMaps)
{
    __shared__ float s[CELLS_PER_MAP];
    __shared__ int   r[CELLS_PER_MAP];
    __shared__ int   c[CELLS_PER_MAP];

    const int map = blockIdx.x;
    const int t   = threadIdx.x;

    if (t < CELLS_PER_MAP) {
        s[t] = candS[map * CELLS_PER_MAP + t];
        r[t] = candR[map * CELLS_PER_MAP + t];
        c[t] = candC[map * CELLS_PER_MAP + t];
    }
    __syncthreads();

    const int topn = *topn_ptr;
    if (t < CELLS_PER_MAP) {
        const float st = s[t];
        int rank = 0;
#pragma unroll 10
        for (int j = 0; j < CELLS_PER_MAP; ++j) {
            const float sj = s[j];
            rank += (sj > st) || (sj == st && j < t);
        }
        if (rank < topn) {
            const size_t sz = (size_t)nMaps * topn;
            out[(size_t)map * topn + rank]          = (float)r[t];
            out[sz + (size_t)map * topn + rank]     = (float)c[t];
            out[2 * sz + (size_t)map * topn + rank] = st;
        }
    }
}

extern "C" void kernel_launch(void* const* d_in, const int* in_sizes, int n_in,
                              void* d_out, int out_size, void* d_ws, size_t ws_size,
                              hipStream_t stream)
{
    const float* maps = (const float*)d_in[0];
    const int*   topn = (const int*)d_in[1];

    const int nMaps  = in_sizes[0] / (MAP_H * MAP_W);   // 128
    const int nCells = nMaps * CELLS_PER_MAP;           // 12800

    float* candS = (float*)d_ws;
    int*   candR = (int*)((char*)d_ws + (size_t)nCells * 4);
    int*   candC = (int*)((char*)d_ws + (size_t)nCells * 8);

    const int blocks1 = (nCells + WAVES_PER_BLOCK - 1) / WAVES_PER_BLOCK;
    patchifier_cell_argmax<<<blocks1, 32 * WAVES_PER_BLOCK, 0, stream>>>(
        maps, candS, candR, candC, nCells);

    patchifier_topk<<<nMaps, 128, 0, stream>>>(candS, candR, candC, topn,
                                               (float*)d_out, nMaps);
}